// ForceFieldTorch_45191645888924
// MI455X (gfx1250) — compile-verified
//
#include <hip/hip_runtime.h>
#include <hip/hip_bf16.h>
#include <math.h>

// ---------------------------------------------------------------------------
// Wave32 reduction via V_WMMA_F32_16X16X4_F32 (CDNA5).
// A-matrix 16x4 layout: lanes 0-15 hold K=0 (v0), K=1 (v1); lanes 16-31 hold
// K=2 (v0), K=3 (v1). With B = all ones:
//   D[m][n] = x(lane m) + x(lane m+16)          (stage 1)
// Summing the 8 D VGPRs per lane leaves the lower-half lanes holding the sum
// over lanes {0..7,16..23} and upper-half lanes over {8..15,24..31}; a second
// identical WMMA adds lower-half + upper-half and broadcasts the full 32-lane
// total to every lane/VGPR.
// ---------------------------------------------------------------------------
typedef __attribute__((ext_vector_type(2))) float v2f;
typedef __attribute__((ext_vector_type(8))) float v8f;

__device__ __forceinline__ float wave_sum32(float x) {
  v2f a; a[0] = x;    a[1] = 0.0f;
  v2f b; b[0] = 1.0f; b[1] = 1.0f;   // B = ones regardless of layout
  v8f c = {};
  v8f d = __builtin_amdgcn_wmma_f32_16x16x4_f32(false, a, false, b,
                                                (short)0, c, false, false);
  float s = ((d[0] + d[1]) + (d[2] + d[3])) + ((d[4] + d[5]) + (d[6] + d[7]));
  v2f a2; a2[0] = s; a2[1] = 0.0f;
  v8f d2 = __builtin_amdgcn_wmma_f32_16x16x4_f32(false, a2, false, b,
                                                 (short)0, c, false, false);
  return d2[0];  // total broadcast to all lanes
}

// Block-level reduction: WMMA wave sums -> LDS -> WMMA again -> 1 atomic/block.
__device__ __forceinline__ void block_acc1(float acc, float* s0,
                                           float* out, int slot) {
  const int lane = threadIdx.x & 31;
  const int wave = threadIdx.x >> 5;
  float t = wave_sum32(acc);
  if (lane == 0) s0[wave] = t;
  __syncthreads();
  if (wave == 0) {                 // wave-uniform branch: EXEC stays all-ones
    float b = wave_sum32(s0[lane]);
    if (lane == 0) atomicAdd(&out[slot], b);
  }
}

__device__ __forceinline__ void block_acc2(float acc0, float acc1,
                                           float* s0, float* s1,
                                           float* out, int slot0, int slot1) {
  const int lane = threadIdx.x & 31;
  const int wave = threadIdx.x >> 5;
  float t0 = wave_sum32(acc0);
  float t1 = wave_sum32(acc1);
  if (lane == 0) { s0[wave] = t0; s1[wave] = t1; }
  __syncthreads();
  if (wave == 0) {
    float b0 = wave_sum32(s0[lane]);
    float b1 = wave_sum32(s1[lane]);
    if (lane == 0) {
      atomicAdd(&out[slot0], b0);
      atomicAdd(&out[slot1], b1);
    }
  }
}

// 12-byte dword-aligned xyz record: lets the backend emit one global_load_b96
// per atom gather instead of three dword loads.
struct __attribute__((packed, aligned(4))) f3 { float x, y, z; };

__device__ __forceinline__ f3 load3(const f3* __restrict__ c, int i) {
  return c[i];
}

// ---------------------------------------------------------------------------

__global__ void k_zero(float* out, int n) {
  int i = blockIdx.x * blockDim.x + threadIdx.x;
  if (i < n) out[i] = 0.0f;
}

__global__ __launch_bounds__(256)
void k_bond(const f3* __restrict__ coords,
            const int* __restrict__ bi, const int* __restrict__ bj,
            const float* __restrict__ bk, const float* __restrict__ br0,
            int n, float* out) {
  __shared__ float s0[32];
  if (threadIdx.x < 32) s0[threadIdx.x] = 0.0f;
  __syncthreads();
  float acc = 0.0f;
  const int stride = gridDim.x * blockDim.x;
  for (int t = blockIdx.x * blockDim.x + threadIdx.x; t < n; t += stride) {
    f3 ci = load3(coords, bi[t]);
    f3 cj = load3(coords, bj[t]);
    float dx = ci.x - cj.x, dy = ci.y - cj.y, dz = ci.z - cj.z;
    float r = sqrtf(dx * dx + dy * dy + dz * dz);
    float d = r - br0[t];
    acc += bk[t] * d * d;
  }
  block_acc1(acc, s0, out, 1);
}

__global__ __launch_bounds__(256)
void k_angle(const f3* __restrict__ coords,
             const int* __restrict__ ai, const int* __restrict__ aj,
             const int* __restrict__ ak,
             const float* __restrict__ k0, const float* __restrict__ t0,
             int n, float* out) {
  __shared__ float s0[32];
  if (threadIdx.x < 32) s0[threadIdx.x] = 0.0f;
  __syncthreads();
  float acc = 0.0f;
  const int stride = gridDim.x * blockDim.x;
  for (int t = blockIdx.x * blockDim.x + threadIdx.x; t < n; t += stride) {
    f3 ci = load3(coords, ai[t]);
    f3 cj = load3(coords, aj[t]);
    f3 ck = load3(coords, ak[t]);
    float ux = ci.x - cj.x, uy = ci.y - cj.y, uz = ci.z - cj.z;
    float vx = ck.x - cj.x, vy = ck.y - cj.y, vz = ck.z - cj.z;
    float uv = ux * vx + uy * vy + uz * vz;
    float uu = ux * ux + uy * uy + uz * uz;
    float vv = vx * vx + vy * vy + vz * vz;
    float ct = uv * rsqrtf(uu * vv);
    ct = fminf(0.999999f, fmaxf(-0.999999f, ct));
    float d = acosf(ct) - t0[t];
    acc += k0[t] * d * d;
  }
  block_acc1(acc, s0, out, 2);
}

__global__ __launch_bounds__(256)
void k_dihed(const f3* __restrict__ coords,
             const int* __restrict__ di, const int* __restrict__ dj,
             const int* __restrict__ dk, const int* __restrict__ dl,
             const float* __restrict__ force, const float* __restrict__ period,
             const float* __restrict__ phase, int n, float* out) {
  __shared__ float s0[32];
  if (threadIdx.x < 32) s0[threadIdx.x] = 0.0f;
  __syncthreads();
  float acc = 0.0f;
  const int stride = gridDim.x * blockDim.x;
  for (int t = blockIdx.x * blockDim.x + threadIdx.x; t < n; t += stride) {
    f3 ci = load3(coords, di[t]);
    f3 cj = load3(coords, dj[t]);
    f3 ck = load3(coords, dk[t]);
    f3 cl = load3(coords, dl[t]);
    float b1x = cj.x - ci.x, b1y = cj.y - ci.y, b1z = cj.z - ci.z;
    float b2x = ck.x - cj.x, b2y = ck.y - cj.y, b2z = ck.z - cj.z;
    float b3x = cl.x - ck.x, b3y = cl.y - ck.y, b3z = cl.z - ck.z;
    // n1 = b1 x b2 ; n2 = b2 x b3
    float n1x = b1y * b2z - b1z * b2y;
    float n1y = b1z * b2x - b1x * b2z;
    float n1z = b1x * b2y - b1y * b2x;
    float n2x = b2y * b3z - b2z * b3y;
    float n2y = b2z * b3x - b2x * b3z;
    float n2z = b2x * b3y - b2y * b3x;
    // cross(n1,n2) . b2_normalized
    float cx = n1y * n2z - n1z * n2y;
    float cy = n1z * n2x - n1x * n2z;
    float cz = n1x * n2y - n1y * n2x;
    float inv_b2 = rsqrtf(b2x * b2x + b2y * b2y + b2z * b2z);
    float y = (cx * b2x + cy * b2y + cz * b2z) * inv_b2;
    float x = n1x * n2x + n1y * n2y + n1z * n2z;
    float phi = atan2f(y, x);
    acc += force[t] * (1.0f + cosf(period[t] * phi - phase[t]));
  }
  block_acc1(acc, s0, out, 3);
}

template <bool S14>
__global__ __launch_bounds__(256)
void k_pairs(const f3* __restrict__ coords, const float* __restrict__ charge,
             const int* __restrict__ atype, const int* __restrict__ nbidx,
             const float* __restrict__ lja, const float* __restrict__ ljb,
             const int* __restrict__ pi, const int* __restrict__ pj,
             const float* __restrict__ scee, const float* __restrict__ scnb,
             const int* __restrict__ ntypes_p,
             int n, float* out, int slotC, int slotL) {
  __shared__ float s0[32], s1[32];
  if (threadIdx.x < 32) { s0[threadIdx.x] = 0.0f; s1[threadIdx.x] = 0.0f; }
  __syncthreads();
  const int nt = *ntypes_p;
  float accC = 0.0f, accL = 0.0f;
  const int stride = gridDim.x * blockDim.x;
  for (int t = blockIdx.x * blockDim.x + threadIdx.x; t < n; t += stride) {
    // Speculative prefetch of next grid-stride tile of the streamed index
    // arrays (global_prefetch_b8; OOB translations are silently dropped).
    __builtin_prefetch(&pi[t + stride], 0, 1);
    __builtin_prefetch(&pj[t + stride], 0, 1);
    int i = pi[t], j = pj[t];
    f3 ci = load3(coords, i);           // global_load_b96 gathers (L2-resident)
    f3 cj = load3(coords, j);
    float dx = ci.x - cj.x, dy = ci.y - cj.y, dz = ci.z - cj.z;
    float r2 = dx * dx + dy * dy + dz * dz;
    float inv_r = rsqrtf(r2);
    float inv_r2 = inv_r * inv_r;
    float r6 = inv_r2 * inv_r2 * inv_r2;
    int idx = nbidx[atype[i] * nt + atype[j]];
    float ec = charge[i] * charge[j] * inv_r;
    float el = lja[idx] * (r6 * r6) - ljb[idx] * r6;
    if (S14) { ec *= scee[t]; el *= scnb[t]; }
    accC += ec;
    accL += el;
  }
  block_acc2(accC, accL, s0, s1, out, slotC, slotL);
}

__global__ void k_finalize(float* out) {
  if (blockIdx.x == 0 && threadIdx.x == 0) {
    float enb = out[5] + out[6] + out[7] + out[8];
    out[9] = enb;
    out[4] = 0.0f;                                   // E_cmap
    out[0] = out[1] + out[2] + out[3] + enb;         // E_total
  }
}

// ---------------------------------------------------------------------------

extern "C" void kernel_launch(void* const* d_in, const int* in_sizes, int n_in,
                              void* d_out, int out_size, void* d_ws, size_t ws_size,
                              hipStream_t stream) {
  const f3*    coords   = (const f3*)d_in[0];
  const float* charge   = (const float*)d_in[1];
  const int*   atype    = (const int*)d_in[2];
  const int*   nbidx    = (const int*)d_in[3];
  const int*   bond_i   = (const int*)d_in[4];
  const int*   bond_j   = (const int*)d_in[5];
  const float* bond_k   = (const float*)d_in[6];
  const float* bond_r0  = (const float*)d_in[7];
  const int*   angle_i  = (const int*)d_in[8];
  const int*   angle_j  = (const int*)d_in[9];
  const int*   angle_k  = (const int*)d_in[10];
  const float* angle_k0 = (const float*)d_in[11];
  const float* angle_t0 = (const float*)d_in[12];
  const int*   dihed_i  = (const int*)d_in[13];
  const int*   dihed_j  = (const int*)d_in[14];
  const int*   dihed_k  = (const int*)d_in[15];
  const int*   dihed_l  = (const int*)d_in[16];
  const float* dihed_f  = (const float*)d_in[17];
  const float* dihed_p  = (const float*)d_in[18];
  const float* dihed_ph = (const float*)d_in[19];
  const float* lja      = (const float*)d_in[20];
  const float* ljb      = (const float*)d_in[21];
  const int*   pair_i   = (const int*)d_in[22];
  const int*   pair_j   = (const int*)d_in[23];
  const int*   p14_i    = (const int*)d_in[24];
  const int*   p14_j    = (const int*)d_in[25];
  const float* scee     = (const float*)d_in[26];
  const float* scnb     = (const float*)d_in[27];
  const int*   ntypes   = (const int*)d_in[28];   // device scalar
  float* out = (float*)d_out;

  const int nb  = in_sizes[4];
  const int na  = in_sizes[8];
  const int nd  = in_sizes[13];
  const int np  = in_sizes[22];
  const int n14 = in_sizes[24];

  auto nblk = [](int n, int per) {
    int b = (n + per - 1) / per;
    return b < 1 ? 1 : b;
  };

  k_zero<<<1, 32, 0, stream>>>(out, out_size);

  k_bond<<<nblk(nb, 256), 256, 0, stream>>>(coords, bond_i, bond_j, bond_k,
                                            bond_r0, nb, out);
  k_angle<<<nblk(na, 256), 256, 0, stream>>>(coords, angle_i, angle_j, angle_k,
                                             angle_k0, angle_t0, na, out);
  k_dihed<<<nblk(nd, 256), 256, 0, stream>>>(coords, dihed_i, dihed_j, dihed_k,
                                             dihed_l, dihed_f, dihed_p,
                                             dihed_ph, nd, out);

  int pb = nblk(np, 256 * 8);          // ~8 pairs/thread grid-stride
  if (pb > 4096) pb = 4096;
  k_pairs<false><<<pb, 256, 0, stream>>>(coords, charge, atype, nbidx, lja, ljb,
                                         pair_i, pair_j, nullptr, nullptr,
                                         ntypes, np, out, 5, 6);
  k_pairs<true><<<nblk(n14, 256), 256, 0, stream>>>(coords, charge, atype, nbidx,
                                                    lja, ljb, p14_i, p14_j,
                                                    scee, scnb, ntypes, n14,
                                                    out, 7, 8);

  k_finalize<<<1, 32, 0, stream>>>(out);
}